// Geo_LoFTREncoderLayer_61426622267549
// MI455X (gfx1250) — compile-verified
//
#include <hip/hip_runtime.h>
#include <math.h>

typedef __attribute__((ext_vector_type(2))) float v2f;
typedef __attribute__((ext_vector_type(8))) float v8f;

#define LQ 1024   // L (queries)
#define SQ 1024   // S (keys)
#define CC 256    // C (channels)
#define HH 8      // heads
#define DD 32     // per-head dim
#define EPSF 1e-5f

// ---------------------------------------------------------------------------
// Kernel 1: Wfpad (256 x 16) = [Wp @ Wp2 | zeros]  (pad N to 16 for WMMA)
// ---------------------------------------------------------------------------
__global__ void k_fuse_wf(const float* __restrict__ Wp,
                          const float* __restrict__ Wp2,
                          float* __restrict__ wfpad) {
    int g = blockIdx.x * blockDim.x + threadIdx.x;      // 0..4095
    if (g >= CC * 16) return;
    int c = g >> 4, n = g & 15;
    float acc = 0.f;
    if (n < HH) {
        for (int k = 0; k < CC; ++k)
            acc += Wp[c * CC + k] * Wp2[k * HH + n];
    }
    wfpad[g] = acc;
}

// ---------------------------------------------------------------------------
// Kernel 2: v = x @ Wv   (1024 x 256)
// ---------------------------------------------------------------------------
__global__ void __launch_bounds__(256)
k_vproj(const float* __restrict__ x, const float* __restrict__ Wv,
        float* __restrict__ vbuf) {
    __shared__ float shx[CC];
    const int row = blockIdx.x, t = threadIdx.x;
    shx[t] = x[row * CC + t];
    __syncthreads();
    float acc = 0.f;
    for (int c = 0; c < CC; ++c) acc += shx[c] * Wv[c * CC + t];
    vbuf[row * CC + t] = acc;
}

// ---------------------------------------------------------------------------
// Kernel 3: per query row l:
//   bias[l, s, h] = embed_qk[l, s, :] . Wf[:, h]   via V_WMMA_F32_16X16X4_F32
//   A[h, l, s]    = softmax_s(bias)                (fused in LDS, 32 KB tile)
// One workgroup (8 waves) per l. Each wave computes 16-row tiles, K=256
// as 64 chained wmma ops. EXEC is all-ones around every WMMA.
// ---------------------------------------------------------------------------
__global__ void __launch_bounds__(256)
k_bias_softmax(const float* __restrict__ E, const float* __restrict__ wfpad,
               float* __restrict__ A) {
    __shared__ float wfs[CC * 16];    // 16 KB padded fused weight
    __shared__ float biasS[SQ * HH];  // 32 KB bias tile for this l
    __shared__ float red[256];        // reduction scratch
    __shared__ float stat[16];        // per-head max / inv-sum

    const int tid = threadIdx.x;
    const int l = blockIdx.x;

    for (int i = tid; i < CC * 16; i += 256) wfs[i] = wfpad[i];
    __syncthreads();

    const int wave = tid >> 5;
    const int lane = tid & 31;
    const int m  = lane & 15;   // M row within tile / N col for B & D
    const int kh = lane >> 4;   // K-half selector

    for (int chunk = wave; chunk < SQ / 16; chunk += 8) {
        const int row0 = chunk * 16;
        const float* erow = E + ((size_t)l * SQ + row0 + m) * CC;
        v8f acc = {};
        #pragma unroll 4
        for (int k = 0; k < CC; k += 4) {
            const float2 av = *reinterpret_cast<const float2*>(erow + k + 2 * kh);
            v2f a; a.x = av.x; a.y = av.y;                 // A[m][k+2kh], A[m][k+2kh+1]
            v2f b;
            b.x = wfs[(k + 2 * kh) * 16 + m];              // B[k+2kh][n]
            b.y = wfs[(k + 1 + 2 * kh) * 16 + m];          // B[k+2kh+1][n]
            acc = __builtin_amdgcn_wmma_f32_16x16x4_f32(
                false, a, false, b, (short)0, acc, false, false);
        }
        if (m < HH) {                                      // only 8 valid head cols
            #pragma unroll
            for (int i = 0; i < 8; ++i)
                biasS[(row0 + i + 8 * kh) * HH + m] = acc[i];
        }
    }
    __syncthreads();

    // ---- softmax over s for each head column (one wave per column) ----
    const int col = tid >> 5;   // 0..7
    const int j   = tid & 31;
    float lmax = -3.4e38f;
    for (int s = j; s < SQ; s += 32) lmax = fmaxf(lmax, biasS[s * HH + col]);
    red[col * 32 + j] = lmax;
    __syncthreads();
    for (int off = 16; off > 0; off >>= 1) {
        if (j < off) red[col*32+j] = fmaxf(red[col*32+j], red[col*32+j+off]);
        __syncthreads();
    }
    if (j == 0) stat[col] = red[col * 32];
    __syncthreads();

    const float cm = stat[col];
    float lsum = 0.f;
    for (int s = j; s < SQ; s += 32) {
        float e = __expf(biasS[s * HH + col] - cm);
        biasS[s * HH + col] = e;
        lsum += e;
    }
    red[col * 32 + j] = lsum;
    __syncthreads();
    for (int off = 16; off > 0; off >>= 1) {
        if (j < off) red[col*32+j] += red[col*32+j+off];
        __syncthreads();
    }
    if (j == 0) stat[8 + col] = 1.0f / red[col * 32];
    __syncthreads();

    // write normalized A in (H, L, S) layout; consecutive tid -> consecutive s
    for (int i = tid; i < SQ * HH; i += 256) {
        int h = i >> 10;            // SQ == 1024
        int s = i & (SQ - 1);
        A[((size_t)h * LQ + l) * SQ + s] = biasS[s * HH + h] * stat[8 + h];
    }
}

// ---------------------------------------------------------------------------
// Kernel 4: msg[l, h*32+d] = sum_s A[h,l,s] * v[s, h*32+d]
// 8 GEMMs (1024x1024)@(1024x32), one 16x16 output tile per wave (WMMA f32).
// A is 32 MB and lives in the 192 MB L2 after kernel 3.
// ---------------------------------------------------------------------------
__global__ void __launch_bounds__(32)
k_attn_msg(const float* __restrict__ A, const float* __restrict__ vbuf,
           float* __restrict__ msg1) {
    const int b  = blockIdx.x;
    const int lt = b & 63;          // L tile (64)
    const int h  = (b >> 6) & 7;    // head (8)
    const int nt = (b >> 9) & 1;    // D half-tile (2)
    const int lane = threadIdx.x;
    const int m  = lane & 15;
    const int kh = lane >> 4;
    const float* Ah = A + ((size_t)h * LQ + lt * 16 + m) * SQ;
    const int colBase = h * DD + nt * 16 + m;

    v8f acc = {};
    #pragma unroll 4
    for (int k = 0; k < SQ; k += 4) {
        const float2 av = *reinterpret_cast<const float2*>(Ah + k + 2 * kh);
        v2f a; a.x = av.x; a.y = av.y;
        v2f bb;
        bb.x = vbuf[(k + 2 * kh) * CC + colBase];
        bb.y = vbuf[(k + 1 + 2 * kh) * CC + colBase];
        acc = __builtin_amdgcn_wmma_f32_16x16x4_f32(
            false, a, false, bb, (short)0, acc, false, false);
    }
    #pragma unroll
    for (int i = 0; i < 8; ++i) {
        const int row = lt * 16 + i + 8 * kh;
        msg1[row * CC + colBase] = acc[i];
    }
}

// ---------------------------------------------------------------------------
// Kernel 5 (fused tail, one block per row):
//   y = msg1 @ Wm ; LN(g1,b1) ; h = [x, y] ; t = relu(h @ W1) ;
//   y2 = t @ W2 ; LN(g2,b2) ; out = x + y2
// ---------------------------------------------------------------------------
__global__ void __launch_bounds__(256)
k_tail(const float* __restrict__ x, const float* __restrict__ msg1,
       const float* __restrict__ Wm, const float* __restrict__ W1,
       const float* __restrict__ W2,
       const float* __restrict__ g1, const float* __restrict__ b1,
       const float* __restrict__ g2, const float* __restrict__ b2,
       float* __restrict__ out) {
    __shared__ float shMsg[CC];
    __shared__ float shH[2 * CC];
    __shared__ float shT[2 * CC];
    __shared__ float red[256];
    const int row = blockIdx.x, t = threadIdx.x;

    shMsg[t] = msg1[row * CC + t];
    const float xv = x[row * CC + t];
    shH[t] = xv;
    __syncthreads();

    float y = 0.f;
    for (int c = 0; c < CC; ++c) y += shMsg[c] * Wm[c * CC + t];

    // LayerNorm 1
    red[t] = y; __syncthreads();
    for (int off = 128; off; off >>= 1) { if (t < off) red[t] += red[t + off]; __syncthreads(); }
    float mu = red[0] * (1.0f / CC); __syncthreads();
    float d = y - mu;
    red[t] = d * d; __syncthreads();
    for (int off = 128; off; off >>= 1) { if (t < off) red[t] += red[t + off]; __syncthreads(); }
    float var = red[0] * (1.0f / CC); __syncthreads();
    shH[CC + t] = d * rsqrtf(var + EPSF) * g1[t] + b1[t];
    __syncthreads();

    // MLP layer 1 (512 -> 512) + ReLU
    #pragma unroll
    for (int jj = 0; jj < 2; ++jj) {
        const int jcol = t + jj * 256;
        float acc = 0.f;
        for (int c = 0; c < 2 * CC; ++c) acc += shH[c] * W1[c * (2 * CC) + jcol];
        shT[jcol] = fmaxf(acc, 0.f);
    }
    __syncthreads();

    // MLP layer 2 (512 -> 256)
    float y2 = 0.f;
    for (int c = 0; c < 2 * CC; ++c) y2 += shT[c] * W2[c * CC + t];

    // LayerNorm 2 + residual
    red[t] = y2; __syncthreads();
    for (int off = 128; off; off >>= 1) { if (t < off) red[t] += red[t + off]; __syncthreads(); }
    mu = red[0] * (1.0f / CC); __syncthreads();
    d = y2 - mu;
    red[t] = d * d; __syncthreads();
    for (int off = 128; off; off >>= 1) { if (t < off) red[t] += red[t + off]; __syncthreads(); }
    var = red[0] * (1.0f / CC);
    out[row * CC + t] = xv + d * rsqrtf(var + EPSF) * g2[t] + b2[t];
}

// ---------------------------------------------------------------------------
extern "C" void kernel_launch(void* const* d_in, const int* in_sizes, int n_in,
                              void* d_out, int out_size, void* d_ws, size_t ws_size,
                              hipStream_t stream) {
    (void)in_sizes; (void)n_in; (void)out_size; (void)ws_size;
    const float* E   = (const float*)d_in[0];   // embed_qk (1,1024,1024,256)
    const float* x   = (const float*)d_in[1];   // (1,1024,256)
    const float* Wv  = (const float*)d_in[2];
    const float* Wp  = (const float*)d_in[3];
    const float* Wp2 = (const float*)d_in[4];
    const float* Wm  = (const float*)d_in[5];
    const float* W1  = (const float*)d_in[6];
    const float* W2  = (const float*)d_in[7];
    const float* g1  = (const float*)d_in[8];
    const float* b1  = (const float*)d_in[9];
    const float* g2  = (const float*)d_in[10];
    const float* b2  = (const float*)d_in[11];
    float* out = (float*)d_out;

    float* ws    = (float*)d_ws;
    float* wfpad = ws;                               // 256*16      =   4096 f
    float* vbuf  = wfpad + CC * 16;                  // 1024*256    = 262144 f
    float* Abuf  = vbuf + (size_t)SQ * CC;           // 8*1024*1024 =  8 Mi f
    float* msg1  = Abuf + (size_t)HH * LQ * SQ;      // 1024*256    = 262144 f

    k_fuse_wf     <<<16,  256, 0, stream>>>(Wp, Wp2, wfpad);
    k_vproj       <<<SQ,  256, 0, stream>>>(x, Wv, vbuf);
    k_bias_softmax<<<LQ,  256, 0, stream>>>(E, wfpad, Abuf);
    k_attn_msg    <<<LQ,   32, 0, stream>>>(Abuf, vbuf, msg1);
    k_tail        <<<LQ,  256, 0, stream>>>(x, msg1, Wm, W1, W2, g1, b1, g2, b2, out);
}